// GNN_31825707664028
// MI455X (gfx1250) — compile-verified
//
#include <hip/hip_runtime.h>
#include <hip/hip_bf16.h>

#define N_NODES   50000
#define N_EDGES   800000
#define IN_DIM    128
#define EMBED     64
#define N_CLASS   10
#define N_GRAPHS  128
#define BN_EPS    1e-5f

typedef __attribute__((ext_vector_type(2))) float v2f;
typedef __attribute__((ext_vector_type(8))) float v8f;

// ---------------------------------------------------------------------------
// WMMA GEMM: C[M x 64] = A[M x K] @ B[K x 64], fp32, row-major.
// One wave computes a 16-row strip across all 64 columns (4 accum tiles).
// B staged TRANSPOSED in LDS with +2 padded stride so each lane's K-pair is
// one aligned ds_load_b64 and 16 lanes land on 16 distinct banks.
// M must be a multiple of 16 (50000 = 3125*16).
// ---------------------------------------------------------------------------
__global__ void __launch_bounds__(256)
gemm_wmma_f32(const float* __restrict__ A, const float* __restrict__ B,
              float* __restrict__ C, int M, int K) {
    __shared__ float Bst[EMBED * (IN_DIM + 2)];   // 64*130 floats = 33.3KB max
    const int tid  = threadIdx.x;
    const int Kpad = K + 2;

    // coalesced global read of B, transposed scatter into LDS
    for (int i = tid; i < K * EMBED; i += blockDim.x) {
        int k = i >> 6, col = i & 63;
        Bst[col * Kpad + k] = B[i];
    }
    __syncthreads();

    const int wavesPerBlock = blockDim.x >> 5;
    const int wave = blockIdx.x * wavesPerBlock + (tid >> 5);
    const int nWaves = M >> 4;
    if (wave >= nWaves) return;

    const int lane = tid & 31;
    const int r    = lane & 15;            // 0..15
    const int half = lane >> 4;            // 0 or 1

    // A row for this lane; lane reads K-pair {kk+2*half, kk+2*half+1}
    const float* arow = A + (size_t)(wave * 16 + r) * K + 2 * half;

    v8f acc[4] = {};

    for (int kk = 0; kk < K; kk += 4) {
        v2f a = *reinterpret_cast<const v2f*>(&arow[kk]);   // global_load_b64
        const int krow = kk + 2 * half;                     // even
#pragma unroll
        for (int t = 0; t < 4; ++t) {
            const int col = t * 16 + r;
            v2f b = *reinterpret_cast<const v2f*>(&Bst[col * Kpad + krow]); // ds_load_b64
            acc[t] = __builtin_amdgcn_wmma_f32_16x16x4_f32(
                false, a, false, b, (short)0, acc[t], false, false);
        }
    }

    // C/D layout: VGPR v -> row (v + 8*half), N = lane%16 within the tile
    const int rowBase = wave * 16 + 8 * half;
#pragma unroll
    for (int t = 0; t < 4; ++t) {
        const int col = t * 16 + r;
#pragma unroll
        for (int v = 0; v < 8; ++v) {
            C[(size_t)(rowBase + v) * EMBED + col] = acc[t][v];
        }
    }
}

// ---------------------------------------------------------------------------
// Graph / elementwise kernels
// ---------------------------------------------------------------------------
__global__ void deg_init_kernel(float* deg, int n) {
    int i = blockIdx.x * blockDim.x + threadIdx.x;
    if (i < n) deg[i] = 1.0f;              // self loop
}

__global__ void deg_count_kernel(const int* __restrict__ dst, float* deg, int e) {
    int i = blockIdx.x * blockDim.x + threadIdx.x;
    if (i < e) atomicAdd(&deg[dst[i]], 1.0f);
}

__global__ void dinv_kernel(float* deg, int n) {
    int i = blockIdx.x * blockDim.x + threadIdx.x;
    if (i < n) deg[i] = rsqrtf(deg[i]);    // in-place: deg -> dinv
}

// agg[n][f..f+3] = b[f..] + xw[n][f..] * dinv[n]^2  (bias + self-loop term)
__global__ void init_agg_kernel(const float* __restrict__ xw,
                                const float* __restrict__ dinv,
                                const float* __restrict__ bias,
                                float* __restrict__ agg, int nq) {   // nq = N*16
    int i = blockIdx.x * blockDim.x + threadIdx.x;
    if (i >= nq) return;
    int n = i >> 4, f = (i & 15) * 4;
    float di = dinv[n];
    float d2 = di * di;
    float4 v = *reinterpret_cast<const float4*>(&xw[(size_t)n * EMBED + f]);
    float4 b = *reinterpret_cast<const float4*>(&bias[f]);
    float4 o;
    o.x = b.x + v.x * d2; o.y = b.y + v.y * d2;
    o.z = b.z + v.z * d2; o.w = b.w + v.w * d2;
    *reinterpret_cast<float4*>(&agg[(size_t)n * EMBED + f]) = o;
}

// 16 threads per edge, float4 gather: agg[dst] += xw[src] * dinv[src]*dinv[dst]
__global__ void edge_agg_kernel(const float* __restrict__ xw,
                                const int* __restrict__ src,
                                const int* __restrict__ dst,
                                const float* __restrict__ dinv,
                                float* __restrict__ agg, int e) {
    int idx = blockIdx.x * blockDim.x + threadIdx.x;
    int ed = idx >> 4;
    if (ed >= e) return;
    int f = (idx & 15) * 4;
    __builtin_prefetch(&src[ed + 4096], 0, 0);   // global_prefetch_b8
    int s = src[ed], d = dst[ed];
    float nrm = dinv[s] * dinv[d];
    float4 v = *reinterpret_cast<const float4*>(&xw[(size_t)s * EMBED + f]);
    float* outp = &agg[(size_t)d * EMBED + f];
    atomicAdd(outp + 0, v.x * nrm);
    atomicAdd(outp + 1, v.y * nrm);
    atomicAdd(outp + 2, v.z * nrm);
    atomicAdd(outp + 3, v.w * nrm);
}

__global__ void zero_kernel(float* p, int n) {
    int i = blockIdx.x * blockDim.x + threadIdx.x;
    if (i < n) p[i] = 0.0f;
}

// Per-column sum / sum-of-squares (BN train stats); 16 threads cover a row
__global__ void colstat_kernel(const float* __restrict__ h,
                               float* __restrict__ sum,
                               float* __restrict__ ssq, int rows) {
    int f = (threadIdx.x & 15) * 4;
    int rowsPerBlk = blockDim.x >> 4;
    int row = blockIdx.x * rowsPerBlk + (threadIdx.x >> 4);
    int stride = gridDim.x * rowsPerBlk;
    float4 s = {0.f, 0.f, 0.f, 0.f}, q = {0.f, 0.f, 0.f, 0.f};
    for (; row < rows; row += stride) {
        float4 v = *reinterpret_cast<const float4*>(&h[(size_t)row * EMBED + f]);
        s.x += v.x; s.y += v.y; s.z += v.z; s.w += v.w;
        q.x += v.x * v.x; q.y += v.y * v.y; q.z += v.z * v.z; q.w += v.w * v.w;
    }
    atomicAdd(&sum[f + 0], s.x); atomicAdd(&sum[f + 1], s.y);
    atomicAdd(&sum[f + 2], s.z); atomicAdd(&sum[f + 3], s.w);
    atomicAdd(&ssq[f + 0], q.x); atomicAdd(&ssq[f + 1], q.y);
    atomicAdd(&ssq[f + 2], q.z); atomicAdd(&ssq[f + 3], q.w);
}

// h = g*(h-mu)*rsqrt(var+eps)+be  (biased var = E[h^2]-mu^2), optional relu
__global__ void bn_apply_kernel(float* __restrict__ h,
                                const float* __restrict__ sum,
                                const float* __restrict__ ssq,
                                const float* __restrict__ g,
                                const float* __restrict__ be,
                                int rows, int nq, int relu) {      // nq = N*16
    int i = blockIdx.x * blockDim.x + threadIdx.x;
    if (i >= nq) return;
    int n = i >> 4, f = (i & 15) * 4;
    float invN = 1.0f / (float)rows;
    float4 sm = *reinterpret_cast<const float4*>(&sum[f]);
    float4 sq = *reinterpret_cast<const float4*>(&ssq[f]);
    float4 gg = *reinterpret_cast<const float4*>(&g[f]);
    float4 bb = *reinterpret_cast<const float4*>(&be[f]);
    float4 v = *reinterpret_cast<float4*>(&h[(size_t)n * EMBED + f]);
    float mu, var;
    mu = sm.x * invN; var = sq.x * invN - mu * mu;
    v.x = gg.x * (v.x - mu) * rsqrtf(var + BN_EPS) + bb.x;
    mu = sm.y * invN; var = sq.y * invN - mu * mu;
    v.y = gg.y * (v.y - mu) * rsqrtf(var + BN_EPS) + bb.y;
    mu = sm.z * invN; var = sq.z * invN - mu * mu;
    v.z = gg.z * (v.z - mu) * rsqrtf(var + BN_EPS) + bb.z;
    mu = sm.w * invN; var = sq.w * invN - mu * mu;
    v.w = gg.w * (v.w - mu) * rsqrtf(var + BN_EPS) + bb.w;
    if (relu) {
        v.x = fmaxf(v.x, 0.f); v.y = fmaxf(v.y, 0.f);
        v.z = fmaxf(v.z, 0.f); v.w = fmaxf(v.w, 0.f);
    }
    *reinterpret_cast<float4*>(&h[(size_t)n * EMBED + f]) = v;
}

__global__ void pool_kernel(const float* __restrict__ h,
                            const int* __restrict__ batch,
                            float* __restrict__ hp, int nq) {       // nq = N*16
    int i = blockIdx.x * blockDim.x + threadIdx.x;
    if (i >= nq) return;
    int n = i >> 4, f = (i & 15) * 4;
    float4 v = *reinterpret_cast<const float4*>(&h[(size_t)n * EMBED + f]);
    float* outp = &hp[(size_t)batch[n] * EMBED + f];
    atomicAdd(outp + 0, v.x);
    atomicAdd(outp + 1, v.y);
    atomicAdd(outp + 2, v.z);
    atomicAdd(outp + 3, v.w);
}

__global__ void fc_kernel(const float* __restrict__ hp,
                          const float* __restrict__ Wfc,
                          const float* __restrict__ bfc,
                          float* __restrict__ out) {
    int i = blockIdx.x * blockDim.x + threadIdx.x;
    if (i >= N_GRAPHS * N_CLASS) return;
    int g = i / N_CLASS, c = i % N_CLASS;
    float s = bfc[c];
#pragma unroll
    for (int f = 0; f < EMBED; ++f)
        s += hp[g * EMBED + f] * Wfc[f * N_CLASS + c];
    out[i] = s;
}

// ---------------------------------------------------------------------------
extern "C" void kernel_launch(void* const* d_in, const int* in_sizes, int n_in,
                              void* d_out, int out_size, void* d_ws, size_t ws_size,
                              hipStream_t stream) {
    const float* x       = (const float*)d_in[0];
    const int*   ei      = (const int*)d_in[1];     // [2, N_EDGES]
    const int*   batch   = (const int*)d_in[2];
    const float* W_embed = (const float*)d_in[3];
    const float* W0      = (const float*)d_in[4];
    const float* b0      = (const float*)d_in[5];
    const float* g0      = (const float*)d_in[6];
    const float* be0     = (const float*)d_in[7];
    const float* W1      = (const float*)d_in[8];
    const float* b1      = (const float*)d_in[9];
    const float* g1      = (const float*)d_in[10];
    const float* be1     = (const float*)d_in[11];
    const float* Wfc     = (const float*)d_in[12];
    const float* bfc     = (const float*)d_in[13];
    float* out = (float*)d_out;

    const int* src = ei;
    const int* dst = ei + N_EDGES;

    // Workspace carve-up (floats)
    float* ws     = (float*)d_ws;
    float* dinv   = ws;                    // 50000
    float* colsum = ws + 50048;            // 64
    float* colss  = ws + 50112;            // 64
    float* hp     = ws + 50176;            // 8192
    float* h      = ws + 65536;            // 3,200,000
    float* xw     = h  + (size_t)N_NODES * EMBED;
    float* agg    = xw + (size_t)N_NODES * EMBED;

    const int NQ = N_NODES * 16;           // float4 granules per feature map
    const int B  = 256;
    const int gNodes = (N_NODES + B - 1) / B;
    const int gNQ    = (NQ + B - 1) / B;
    const int gEdge  = (N_EDGES + B - 1) / B;
    const int gEdgeF = ((size_t)N_EDGES * 16 + B - 1) / B;
    const int gGemm  = ((N_NODES / 16) + (B / 32) - 1) / (B / 32);  // 3125 waves

    // --- degrees -> dinv ---
    deg_init_kernel<<<gNodes, B, 0, stream>>>(dinv, N_NODES);
    deg_count_kernel<<<gEdge, B, 0, stream>>>(dst, dinv, N_EDGES);
    dinv_kernel<<<gNodes, B, 0, stream>>>(dinv, N_NODES);

    // --- node embedding: h = x @ W_embed ---
    gemm_wmma_f32<<<gGemm, B, 0, stream>>>(x, W_embed, h, N_NODES, IN_DIM);

    // --- GCN layer 0 ---
    gemm_wmma_f32<<<gGemm, B, 0, stream>>>(h, W0, xw, N_NODES, EMBED);
    init_agg_kernel<<<gNQ, B, 0, stream>>>(xw, dinv, b0, agg, NQ);
    edge_agg_kernel<<<gEdgeF, B, 0, stream>>>(xw, src, dst, dinv, agg, N_EDGES);
    zero_kernel<<<1, 128, 0, stream>>>(colsum, 128);   // colsum + colss contiguous
    colstat_kernel<<<256, B, 0, stream>>>(agg, colsum, colss, N_NODES);
    bn_apply_kernel<<<gNQ, B, 0, stream>>>(agg, colsum, colss, g0, be0,
                                           N_NODES, NQ, /*relu=*/1);

    // --- GCN layer 1 (agg is the new h; reuse h buffer for the new agg) ---
    gemm_wmma_f32<<<gGemm, B, 0, stream>>>(agg, W1, xw, N_NODES, EMBED);
    init_agg_kernel<<<gNQ, B, 0, stream>>>(xw, dinv, b1, h, NQ);
    edge_agg_kernel<<<gEdgeF, B, 0, stream>>>(xw, src, dst, dinv, h, N_EDGES);
    zero_kernel<<<1, 128, 0, stream>>>(colsum, 128);
    colstat_kernel<<<256, B, 0, stream>>>(h, colsum, colss, N_NODES);
    bn_apply_kernel<<<gNQ, B, 0, stream>>>(h, colsum, colss, g1, be1,
                                           N_NODES, NQ, /*relu=*/0);

    // --- global add pool + FC ---
    zero_kernel<<<(N_GRAPHS * EMBED + B - 1) / B, B, 0, stream>>>(hp, N_GRAPHS * EMBED);
    pool_kernel<<<gNQ, B, 0, stream>>>(h, batch, hp, NQ);
    fc_kernel<<<(N_GRAPHS * N_CLASS + B - 1) / B, B, 0, stream>>>(hp, Wfc, bfc, out);
}